// GCN2_55920474193958
// MI455X (gfx1250) — compile-verified
//
#include <hip/hip_runtime.h>
#include <hip/hip_bf16.h>
#include <math.h>

typedef float v2f __attribute__((ext_vector_type(2)));
typedef float v8f __attribute__((ext_vector_type(8)));

#define NFEAT 256
#define NHID  128
#define NCLASS 40
#define NCPAD  48      // NCLASS padded to multiple of 16
#define MT     5       // M-tiles (16 rows each) per wave; 100000/(16*5)=1250 exact

// ---------------------------------------------------------------------------
// Degree / normalization
// ---------------------------------------------------------------------------
__global__ void gcn_deg_init(float* __restrict__ deg, int n) {
    int i = blockIdx.x * blockDim.x + threadIdx.x;
    if (i < n) deg[i] = 1.0f;   // self-loop
}

__global__ void gcn_deg_accum(const int* __restrict__ dst, float* __restrict__ deg, int e) {
    int i = blockIdx.x * blockDim.x + threadIdx.x;
    if (i < e) atomicAdd(&deg[dst[i]], 1.0f);
}

__global__ void gcn_deg_finalize(float* __restrict__ dis, int n) {
    int i = blockIdx.x * blockDim.x + threadIdx.x;
    if (i < n) dis[i] = rsqrtf(fmaxf(dis[i], 1.0f));   // deg >= 1 already
}

// ---------------------------------------------------------------------------
// Pack W [K x NC] into pair-interleaved, column-padded form:
//   Wp[(k>>1)*2*LDB + c*2 + (k&1)] = (c < NC) ? W[k*NC + c] : 0
// so a B fragment (rows ka, ka+1 at column c) is ONE contiguous b64 load.
// ---------------------------------------------------------------------------
__global__ void gcn_pack_w(const float* __restrict__ W, float* __restrict__ Wp,
                           int K, int NC, int ldb)
{
    int i = blockIdx.x * blockDim.x + threadIdx.x;   // over K*ldb
    if (i >= K * ldb) return;
    int k = i / ldb;
    int c = i - k * ldb;
    float v = (c < NC) ? W[k * NC + c] : 0.0f;
    Wp[(k >> 1) * (2 * ldb) + c * 2 + (k & 1)] = v;
}

// ---------------------------------------------------------------------------
// Dense GEMM: C[M x NCOUT] = A[M x K] @ W[K x *] via fp32 WMMA 16x16x4.
// One wave computes MT stacked 16x16 tiles (80 rows x 16 cols): each B
// fragment (one b64 load from packed weights) feeds MT WMMAs.
// K / LDB / NCOUT are compile-time, so every in-loop address is
// base + immediate, expression-addressed from the kernarg pointers
// (global_load, partial s_wait_loadcnt legal). Loads grouped before WMMAs.
// ---------------------------------------------------------------------------
template<int K, int LDB, int NCOUT>
__global__ __launch_bounds__(32) void gcn_gemm_wmma_f32(
    const float* __restrict__ A, const float* __restrict__ Wp,
    float* __restrict__ C)
{
    const int lane = threadIdx.x;
    const int half = lane >> 4;
    const int r    = lane & 15;

    const int rowBase = blockIdx.x * (16 * MT);
    const int bcol    = blockIdx.y * 16 + r;

    // A: row = rowBase + 16*t + r, element k + 2*half
    const float* abase = A + (size_t)(rowBase + r) * K + 2 * half;
    // B packed: pair p = k/2 + half  ->  Wp + p*2*LDB + 2*bcol
    const float* bbase = Wp + (size_t)half * (2 * LDB) + 2 * bcol;

    v8f acc[MT];
    #pragma unroll
    for (int t = 0; t < MT; ++t) acc[t] = (v8f){};

    #pragma unroll 4
    for (int k = 0; k < K; k += 4) {
        // ---- load group: 1 B fragment + MT A fragments (all b64) ----
        v2f b = *reinterpret_cast<const v2f*>(bbase + (size_t)k * LDB);
        v2f a[MT];
        #pragma unroll
        for (int t = 0; t < MT; ++t)
            a[t] = *reinterpret_cast<const v2f*>(abase + (size_t)t * (16 * K) + k);
        // ---- compute group: MT WMMAs reuse the same B fragment ----
        #pragma unroll
        for (int t = 0; t < MT; ++t)
            acc[t] = __builtin_amdgcn_wmma_f32_16x16x4_f32(
                /*neg_a=*/false, a[t], /*neg_b=*/false, b,
                /*c_mod=*/(short)0, acc[t], /*reuse_a=*/false, /*reuse_b=*/false);
    }

    // D layout: VGPR v, lane l -> row = v + 8*half, col = lane&15
    if (bcol < NCOUT) {
        #pragma unroll
        for (int t = 0; t < MT; ++t) {
            float* crow = C + (size_t)(rowBase + 16 * t + 8 * half) * NCOUT + bcol;
            #pragma unroll
            for (int v = 0; v < 8; ++v)
                crow[(size_t)v * NCOUT] = acc[t][v];
        }
    }
}

// ---------------------------------------------------------------------------
// out[i][f..f+3] = dis[i]^2 * h[i][f..f+3] + b[f..f+3]   (self-loop + bias)
// grid: (ceil(n/256), NC/4)
// ---------------------------------------------------------------------------
__global__ void gcn_selfloop_init(const float* __restrict__ h,
                                  const float* __restrict__ dis,
                                  const float* __restrict__ bias,
                                  float* __restrict__ out, int n, int nc)
{
    int i = blockIdx.x * blockDim.x + threadIdx.x;
    if (i >= n) return;
    int f = blockIdx.y * 4;
    float d = dis[i];
    float w = d * d;
    float4 hv = *reinterpret_cast<const float4*>(h + (size_t)i * nc + f);
    float4 bv = *reinterpret_cast<const float4*>(bias + f);
    float4 o;
    o.x = w * hv.x + bv.x;
    o.y = w * hv.y + bv.y;
    o.z = w * hv.z + bv.z;
    o.w = w * hv.w + bv.w;
    *reinterpret_cast<float4*>(out + (size_t)i * nc + f) = o;
}

// ---------------------------------------------------------------------------
// out[dst[e]][f..f+3] += dis[src]*dis[dst] * h[src][f..f+3]
// grid: (ceil(E/256), NC/4); atomics resolve in L2 (h fits in 192MB L2)
// ---------------------------------------------------------------------------
__global__ void gcn_edge_agg(const int* __restrict__ src, const int* __restrict__ dst,
                             const float* __restrict__ dis, const float* __restrict__ h,
                             float* __restrict__ out, int e, int nc)
{
    int i = blockIdx.x * blockDim.x + threadIdx.x;
    if (i >= e) return;
    int f = blockIdx.y * 4;
    int s = src[i];
    int d = dst[i];
    float nrm = dis[s] * dis[d];
    float4 hv = *reinterpret_cast<const float4*>(h + (size_t)s * nc + f);
    float* o = out + (size_t)d * nc + f;
    atomicAdd(o + 0, nrm * hv.x);
    atomicAdd(o + 1, nrm * hv.y);
    atomicAdd(o + 2, nrm * hv.z);
    atomicAdd(o + 3, nrm * hv.w);
}

// ---------------------------------------------------------------------------
// In-place ReLU over n4 float4 elements
// ---------------------------------------------------------------------------
__global__ void gcn_relu4(float4* __restrict__ x, int n4) {
    int i = blockIdx.x * blockDim.x + threadIdx.x;
    if (i >= n4) return;
    float4 v = x[i];
    v.x = fmaxf(v.x, 0.0f);
    v.y = fmaxf(v.y, 0.0f);
    v.z = fmaxf(v.z, 0.0f);
    v.w = fmaxf(v.w, 0.0f);
    x[i] = v;
}

// ---------------------------------------------------------------------------
// h[idx_com[j]][f..f+3] = hist[idx_data[j]][f..f+3]
// grid: (ceil(m/256), NC/4)
// ---------------------------------------------------------------------------
__global__ void gcn_overwrite(float* __restrict__ h, const float* __restrict__ hist,
                              const int* __restrict__ idx_com,
                              const int* __restrict__ idx_data, int m, int nc)
{
    int j = blockIdx.x * blockDim.x + threadIdx.x;
    if (j >= m) return;
    int f = blockIdx.y * 4;
    int ci = idx_com[j];
    int di = idx_data[j];
    *reinterpret_cast<float4*>(h + (size_t)ci * nc + f) =
        *reinterpret_cast<const float4*>(hist + (size_t)di * nc + f);
}

// ---------------------------------------------------------------------------
// In-place per-row log_softmax over NC=40 columns (row = 160B, contiguous)
// ---------------------------------------------------------------------------
__global__ void gcn_log_softmax(float* __restrict__ out, int n, int nc) {
    int i = blockIdx.x * blockDim.x + threadIdx.x;
    if (i >= n) return;
    float* row = out + (size_t)i * nc;
    float m = -INFINITY;
    for (int c = 0; c < nc; ++c) m = fmaxf(m, row[c]);
    float s = 0.0f;
    for (int c = 0; c < nc; ++c) s += expf(row[c] - m);
    float ls = logf(s);
    for (int c = 0; c < nc; ++c) row[c] = row[c] - m - ls;
}

// ---------------------------------------------------------------------------
// Host orchestration
// ---------------------------------------------------------------------------
extern "C" void kernel_launch(void* const* d_in, const int* in_sizes, int n_in,
                              void* d_out, int out_size, void* d_ws, size_t ws_size,
                              hipStream_t stream)
{
    const float* x     = (const float*)d_in[0];
    const float* W0    = (const float*)d_in[1];
    const float* b0    = (const float*)d_in[2];
    const float* W1    = (const float*)d_in[3];
    const float* b1    = (const float*)d_in[4];
    const float* W2    = (const float*)d_in[5];
    const float* b2    = (const float*)d_in[6];
    const float* hist0 = (const float*)d_in[7];
    const float* hist1 = (const float*)d_in[8];
    const int* esrc    = (const int*)d_in[9];
    const int* edst    = (const int*)d_in[10];
    const int* idxd0   = (const int*)d_in[11];
    const int* idxc0   = (const int*)d_in[12];
    const int* idxd1   = (const int*)d_in[13];
    const int* idxc1   = (const int*)d_in[14];

    const int N  = in_sizes[0] / NFEAT;
    const int E  = in_sizes[9];
    const int M0 = in_sizes[11];
    const int M1 = in_sizes[13];

    float* dis  = (float*)d_ws;                    // N
    float* bufA = dis + N;                         // N * NHID  (pre-aggregation h)
    float* bufB = bufA + (size_t)N * NHID;         // N * NHID  (layer output)
    float* W0p  = bufB + (size_t)N * NHID;         // NFEAT * NHID  (packed)
    float* W1p  = W0p + (size_t)NFEAT * NHID;      // NHID * NHID   (packed)
    float* W2p  = W1p + (size_t)NHID * NHID;       // NHID * NCPAD  (packed+padded)
    float* fout = (float*)d_out;                   // N * NCLASS

    const int TB = 256;
    dim3 blk1(TB);

    // --- degrees / normalization + weight packing --------------------------
    gcn_deg_init    <<<dim3((N + TB - 1) / TB), blk1, 0, stream>>>(dis, N);
    gcn_deg_accum   <<<dim3((E + TB - 1) / TB), blk1, 0, stream>>>(edst, dis, E);
    gcn_deg_finalize<<<dim3((N + TB - 1) / TB), blk1, 0, stream>>>(dis, N);
    gcn_pack_w<<<dim3((NFEAT * NHID + TB - 1) / TB), blk1, 0, stream>>>(
        W0, W0p, NFEAT, NHID, NHID);
    gcn_pack_w<<<dim3((NHID * NHID + TB - 1) / TB), blk1, 0, stream>>>(
        W1, W1p, NHID, NHID, NHID);
    gcn_pack_w<<<dim3((NHID * NCPAD + TB - 1) / TB), blk1, 0, stream>>>(
        W2, W2p, NHID, NCLASS, NCPAD);

    // --- layer 0: h = relu(Ahat @ (x@W0) + b0); overwrite hist0 ------------
    gcn_gemm_wmma_f32<NFEAT, NHID, NHID>
        <<<dim3(N / (16 * MT), NHID / 16), dim3(32), 0, stream>>>(x, W0p, bufA);
    gcn_selfloop_init<<<dim3((N + TB - 1) / TB, NHID / 4), blk1, 0, stream>>>(
        bufA, dis, b0, bufB, N, NHID);
    gcn_edge_agg<<<dim3((E + TB - 1) / TB, NHID / 4), blk1, 0, stream>>>(
        esrc, edst, dis, bufA, bufB, E, NHID);
    gcn_relu4<<<dim3(((N * (NHID / 4)) + TB - 1) / TB), blk1, 0, stream>>>(
        (float4*)bufB, N * (NHID / 4));
    gcn_overwrite<<<dim3((M0 + TB - 1) / TB, NHID / 4), blk1, 0, stream>>>(
        bufB, hist0, idxc0, idxd0, M0, NHID);

    // --- layer 1: h = relu(Ahat @ (h@W1) + b1); overwrite hist1 ------------
    gcn_gemm_wmma_f32<NHID, NHID, NHID>
        <<<dim3(N / (16 * MT), NHID / 16), dim3(32), 0, stream>>>(bufB, W1p, bufA);
    gcn_selfloop_init<<<dim3((N + TB - 1) / TB, NHID / 4), blk1, 0, stream>>>(
        bufA, dis, b1, bufB, N, NHID);
    gcn_edge_agg<<<dim3((E + TB - 1) / TB, NHID / 4), blk1, 0, stream>>>(
        esrc, edst, dis, bufA, bufB, E, NHID);
    gcn_relu4<<<dim3(((N * (NHID / 4)) + TB - 1) / TB), blk1, 0, stream>>>(
        (float4*)bufB, N * (NHID / 4));
    gcn_overwrite<<<dim3((M1 + TB - 1) / TB, NHID / 4), blk1, 0, stream>>>(
        bufB, hist1, idxc1, idxd1, M1, NHID);

    // --- final layer: out = log_softmax(Ahat @ (h@W2) + b2) ----------------
    gcn_gemm_wmma_f32<NHID, NCPAD, NCLASS>
        <<<dim3(N / (16 * MT), NCPAD / 16), dim3(32), 0, stream>>>(bufB, W2p, bufA);
    gcn_selfloop_init<<<dim3((N + TB - 1) / TB, NCLASS / 4), blk1, 0, stream>>>(
        bufA, dis, b2, fout, N, NCLASS);
    gcn_edge_agg<<<dim3((E + TB - 1) / TB, NCLASS / 4), blk1, 0, stream>>>(
        esrc, edst, dis, bufA, fout, E, NCLASS);
    gcn_log_softmax<<<dim3((N + TB - 1) / TB), blk1, 0, stream>>>(fout, N, NCLASS);
}